// MoEFeedForward_6476810683035
// MI455X (gfx1250) — compile-verified
//
#include <hip/hip_runtime.h>
#include <hip/hip_bf16.h>

typedef float v2f __attribute__((ext_vector_type(2)));
typedef float v8f __attribute__((ext_vector_type(8)));

// ---------------- problem constants ----------------
constexpr int BT   = 4 * 2048;   // tokens
constexpr int D    = 1024;       // hidden dim
constexpr int E    = 16;         // experts
constexpr int F    = 768;        // ffn dim
constexpr int TOPK = 2;
constexpr int TM   = 32;                         // rows per GEMM tile
constexpr int MAXTILES = (BT * TOPK) / TM + E;   // 528 (worst-case padding)
constexpr int XS = D + 4;        // LDS row stride for X (bank-conflict pad)
constexpr int HS = F + 4;        // LDS row stride for H
constexpr size_t SMEM_BYTES = (size_t)(TM * XS + TM * HS) * 4; // 230400 B

// ---------------- helpers ----------------
__device__ __forceinline__ v8f wmma4(v2f a, v2f b, v8f c) {
  return __builtin_amdgcn_wmma_f32_16x16x4_f32(false, a, false, b, (short)0, c,
                                               false, false);
}

__device__ __forceinline__ float fast_silu(float g) {
  // g * sigmoid(g) with v_rcp_f32 instead of IEEE division
  return g * __builtin_amdgcn_rcpf(1.f + __expf(-g));
}

// ---------------- init kernels ----------------
__global__ void zero_out_kernel(float4* __restrict__ p, int n4) {
  int i = blockIdx.x * blockDim.x + threadIdx.x;
  int stride = gridDim.x * blockDim.x;
  float4 z; z.x = 0.f; z.y = 0.f; z.z = 0.f; z.w = 0.f;
  for (; i < n4; i += stride) p[i] = z;
}

__global__ void init_ctrl_kernel(int* cnt, int* fill, float* psum) {
  int i = threadIdx.x;
  if (i < E) { cnt[i] = 0; fill[i] = 0; psum[i] = 0.f; }
}

// ---------------- router: one wave32 per token ----------------
__global__ void router_kernel(const float* __restrict__ hidden,
                              const float* __restrict__ rw,
                              int* __restrict__ tokidx, float* __restrict__ tokw,
                              int* __restrict__ cnt, float* __restrict__ psum) {
  int wave = (blockIdx.x * blockDim.x + threadIdx.x) >> 5;
  int lane = threadIdx.x & 31;
  if (wave >= BT) return;
  const float* h = hidden + (size_t)wave * D;

  float acc[E];
#pragma unroll
  for (int e = 0; e < E; ++e) acc[e] = 0.f;

  for (int d = lane; d < D; d += 32) {
    float hv = h[d];
    const float4* r4 = reinterpret_cast<const float4*>(rw + (size_t)d * E);
#pragma unroll
    for (int q = 0; q < 4; ++q) {
      float4 rv = r4[q];
      acc[4 * q + 0] += hv * rv.x;
      acc[4 * q + 1] += hv * rv.y;
      acc[4 * q + 2] += hv * rv.z;
      acc[4 * q + 3] += hv * rv.w;
    }
  }
  // cross-lane reduction (wave32)
#pragma unroll
  for (int off = 16; off >= 1; off >>= 1) {
#pragma unroll
    for (int e = 0; e < E; ++e) acc[e] += __shfl_xor(acc[e], off, 32);
  }

  if (lane == 0) {
    float m = acc[0];
#pragma unroll
    for (int e = 1; e < E; ++e) m = fmaxf(m, acc[e]);
    float p[E];
    float s = 0.f;
#pragma unroll
    for (int e = 0; e < E; ++e) { p[e] = __expf(acc[e] - m); s += p[e]; }
    float inv = 1.f / s;   // one precise division per token, fine
#pragma unroll
    for (int e = 0; e < E; ++e) p[e] *= inv;

    int i1 = 0; float p1 = p[0];
#pragma unroll
    for (int e = 1; e < E; ++e) if (p[e] > p1) { p1 = p[e]; i1 = e; }
    int i2 = (i1 == 0) ? 1 : 0; float p2 = p[i2];
#pragma unroll
    for (int e = 0; e < E; ++e)
      if (e != i1 && p[e] > p2) { p2 = p[e]; i2 = e; }

    float denom = p1 + p2;
    if (denom < 1e-9f) denom = 1e-9f;
    float invd = 1.f / denom;
    tokidx[2 * wave + 0] = i1;
    tokidx[2 * wave + 1] = i2;
    tokw[2 * wave + 0] = p1 * invd;
    tokw[2 * wave + 1] = p2 * invd;
    atomicAdd(&cnt[i1], 1);
    atomicAdd(&cnt[i2], 1);
#pragma unroll
    for (int e = 0; e < E; ++e) atomicAdd(&psum[e], p[e]);
  }
}

// ---------------- segment setup (serial, tiny) ----------------
__global__ void segment_kernel(const int* __restrict__ cnt, int* __restrict__ seg,
                               int* __restrict__ texp, int* __restrict__ rtok) {
  if (threadIdx.x != 0 || blockIdx.x != 0) return;
  int run = 0;
  for (int e = 0; e < E; ++e) {
    seg[e] = run;
    int c = cnt[e];
    int pad = ((c + TM - 1) / TM) * TM;
    for (int q = c; q < pad; ++q) rtok[run + q] = -1;  // padding rows
    run += pad;
  }
  seg[E] = run;
  int ntiles = run / TM;
  for (int m = 0; m < MAXTILES; ++m) {
    if (m < ntiles) {
      int s = m * TM;
      int e = 0;
      while (!(s >= seg[e] && s < seg[e + 1])) ++e;
      texp[m] = e;
    } else {
      texp[m] = -1;
    }
  }
}

// ---------------- scatter tokens into expert segments ----------------
__global__ void scatter_kernel(const int* __restrict__ tokidx,
                               const float* __restrict__ tokw,
                               const int* __restrict__ seg, int* __restrict__ fill,
                               int* __restrict__ rtok, float* __restrict__ rwgt) {
  int i = blockIdx.x * blockDim.x + threadIdx.x;
  if (i >= BT * TOPK) return;
  int e = tokidx[i];
  int pos = seg[e] + atomicAdd(&fill[e], 1);
  rtok[pos] = i >> 1;   // token id
  rwgt[pos] = tokw[i];
}

// ---------------- fused expert FFN: 32 tokens x (gate,up,down) ----------------
__global__ __launch_bounds__(512)
void ffn_kernel(const float* __restrict__ hidden, const float* __restrict__ Wg_,
                const float* __restrict__ Wu_, const float* __restrict__ Wd_,
                const int* __restrict__ rtok, const float* __restrict__ rwgt,
                const int* __restrict__ texp, float* __restrict__ out) {
  extern __shared__ float smem[];
  float* Xs = smem;               // [TM][XS]
  float* Hsh = smem + TM * XS;    // [TM][HS]

  const int m = blockIdx.x;
  const int e = texp[m];
  if (e < 0) return;

  const int tid  = threadIdx.x;
  const int lane = tid & 31;
  const int w    = tid >> 5;      // wave id, 0..15
  const int l15  = lane & 15;
  const int lh   = lane >> 4;     // 0 or 1 (lane half)

  // ---- stage X tile (32 x 1024) into LDS; pad rows -> zeros ----
  {
    int r = tid >> 4;        // 0..31
    int c = tid & 15;        // 16 threads per row
    int t = rtok[m * TM + r];
    const float4* src =
        (t >= 0) ? reinterpret_cast<const float4*>(hidden + (size_t)t * D) : nullptr;
    float4 z; z.x = 0.f; z.y = 0.f; z.z = 0.f; z.w = 0.f;
#pragma unroll
    for (int j = 0; j < 16; ++j) {
      int k4 = c + j * 16;   // float4 index within row (0..255)
      float4 v = (t >= 0) ? src[k4] : z;
      *reinterpret_cast<float4*>(&Xs[r * XS + k4 * 4]) = v;
    }
  }
  __syncthreads();

  const float* Wg = Wg_ + (size_t)e * D * F;
  const float* Wu = Wu_ + (size_t)e * D * F;
  const float* Wd = Wd_ + (size_t)e * F * D;

  v8f z8 = {0.f, 0.f, 0.f, 0.f, 0.f, 0.f, 0.f, 0.f};
  const int cb1 = w * 48 + l15;   // this wave's F-column base (phases 1-2)

  // ---- phase 1: gate = X @ Wg ; H = silu(gate) ----
  {
    v8f acc[2][3];
#pragma unroll
    for (int mt = 0; mt < 2; ++mt)
#pragma unroll
      for (int nt = 0; nt < 3; ++nt) acc[mt][nt] = z8;

#pragma unroll 4
    for (int kk = 0; kk < D; kk += 4) {
      int k = kk + 2 * lh;
      v2f a0, a1;
      a0.x = Xs[l15 * XS + k];        a0.y = Xs[l15 * XS + k + 1];
      a1.x = Xs[(l15 + 16) * XS + k]; a1.y = Xs[(l15 + 16) * XS + k + 1];
#pragma unroll
      for (int nt = 0; nt < 3; ++nt) {
        int cbn = cb1 + nt * 16;
        v2f b;
        b.x = Wg[(size_t)k * F + cbn];
        b.y = Wg[(size_t)(k + 1) * F + cbn];
        acc[0][nt] = wmma4(a0, b, acc[0][nt]);
        acc[1][nt] = wmma4(a1, b, acc[1][nt]);
      }
    }
#pragma unroll
    for (int mt = 0; mt < 2; ++mt)
#pragma unroll
      for (int nt = 0; nt < 3; ++nt)
#pragma unroll
        for (int r = 0; r < 8; ++r) {
          int row = mt * 16 + lh * 8 + r;
          Hsh[row * HS + cb1 + nt * 16] = fast_silu(acc[mt][nt][r]);
        }
  }

  // ---- phase 2: up = X @ Wu ; H *= up (wave-private H columns) ----
  {
    v8f acc[2][3];
#pragma unroll
    for (int mt = 0; mt < 2; ++mt)
#pragma unroll
      for (int nt = 0; nt < 3; ++nt) acc[mt][nt] = z8;

#pragma unroll 4
    for (int kk = 0; kk < D; kk += 4) {
      int k = kk + 2 * lh;
      v2f a0, a1;
      a0.x = Xs[l15 * XS + k];        a0.y = Xs[l15 * XS + k + 1];
      a1.x = Xs[(l15 + 16) * XS + k]; a1.y = Xs[(l15 + 16) * XS + k + 1];
#pragma unroll
      for (int nt = 0; nt < 3; ++nt) {
        int cbn = cb1 + nt * 16;
        v2f b;
        b.x = Wu[(size_t)k * F + cbn];
        b.y = Wu[(size_t)(k + 1) * F + cbn];
        acc[0][nt] = wmma4(a0, b, acc[0][nt]);
        acc[1][nt] = wmma4(a1, b, acc[1][nt]);
      }
    }
#pragma unroll
    for (int mt = 0; mt < 2; ++mt)
#pragma unroll
      for (int nt = 0; nt < 3; ++nt)
#pragma unroll
        for (int r = 0; r < 8; ++r) {
          int row = mt * 16 + lh * 8 + r;
          Hsh[row * HS + cb1 + nt * 16] *= acc[mt][nt][r];
        }
  }
  __syncthreads();

  // ---- phase 3: Y = H @ Wd ; weighted atomic scatter into out ----
  {
    v8f acc[2][4];
#pragma unroll
    for (int mt = 0; mt < 2; ++mt)
#pragma unroll
      for (int nt = 0; nt < 4; ++nt) acc[mt][nt] = z8;

    const int cb3 = w * 64 + l15;   // this wave's D-column base
#pragma unroll 4
    for (int kk = 0; kk < F; kk += 4) {
      int k = kk + 2 * lh;
      v2f a0, a1;
      a0.x = Hsh[l15 * HS + k];        a0.y = Hsh[l15 * HS + k + 1];
      a1.x = Hsh[(l15 + 16) * HS + k]; a1.y = Hsh[(l15 + 16) * HS + k + 1];
#pragma unroll
      for (int nt = 0; nt < 4; ++nt) {
        int cbn = cb3 + nt * 16;
        v2f b;
        b.x = Wd[(size_t)k * D + cbn];
        b.y = Wd[(size_t)(k + 1) * D + cbn];
        acc[0][nt] = wmma4(a0, b, acc[0][nt]);
        acc[1][nt] = wmma4(a1, b, acc[1][nt]);
      }
    }

#pragma unroll
    for (int mt = 0; mt < 2; ++mt) {
#pragma unroll
      for (int r = 0; r < 8; ++r) {
        int row = mt * 16 + lh * 8 + r;
        int t = rtok[m * TM + row];
        if (t < 0) continue;
        float wt = rwgt[m * TM + row];
        float* orow = out + (size_t)t * D;
#pragma unroll
        for (int nt = 0; nt < 4; ++nt)
          atomicAdd(&orow[cb3 + nt * 16], wt * acc[mt][nt][r]);
      }
    }
  }
}

// ---------------- aux loss ----------------
__global__ void aux_kernel(const int* __restrict__ cnt,
                           const float* __restrict__ psum, float* out_aux) {
  if (threadIdx.x == 0 && blockIdx.x == 0) {
    float s = 0.f;
    for (int e = 0; e < E; ++e) s += (float)cnt[e] * psum[e];
    out_aux[0] = 0.001f * (float)E * s / ((float)BT * (float)BT);
  }
}

// ---------------- launch ----------------
extern "C" void kernel_launch(void* const* d_in, const int* in_sizes, int n_in,
                              void* d_out, int out_size, void* d_ws, size_t ws_size,
                              hipStream_t stream) {
  const float* hidden = (const float*)d_in[0];
  const float* gate_p = (const float*)d_in[1];
  const float* up_p   = (const float*)d_in[2];
  const float* down_p = (const float*)d_in[3];
  const float* rw     = (const float*)d_in[4];
  float* out = (float*)d_out;

  // workspace layout (4-byte words)
  int*   cnt    = (int*)d_ws;                       // 16
  int*   fill   = cnt + 16;                         // 16
  int*   seg    = fill + 16;                        // 17
  float* psum   = (float*)(seg + 32);               // 16  (seg padded to 32)
  int*   texp   = (int*)(psum + 16);                // MAXTILES
  int*   tokidx = texp + MAXTILES;                  // BT*TOPK
  float* tokw   = (float*)(tokidx + BT * TOPK);     // BT*TOPK
  int*   rtok   = (int*)(tokw + BT * TOPK);         // BT*TOPK + E*TM
  float* rwgt   = (float*)(rtok + BT * TOPK + E * TM);

  // 1. zero output (we atomically accumulate into it)
  zero_out_kernel<<<2048, 256, 0, stream>>>((float4*)out, (BT * D) / 4);
  init_ctrl_kernel<<<1, 64, 0, stream>>>(cnt, fill, psum);

  // 2. router + softmax + top-2
  router_kernel<<<BT / 8, 256, 0, stream>>>(hidden, rw, tokidx, tokw, cnt, psum);

  // 3. expert segments (padded to 32-row tiles)
  segment_kernel<<<1, 32, 0, stream>>>(cnt, seg, texp, rtok);

  // 4. scatter tokens to segments
  scatter_kernel<<<(BT * TOPK + 255) / 256, 256, 0, stream>>>(tokidx, tokw, seg,
                                                              fill, rtok, rwgt);

  // 5. fused expert FFN (WMMA f32)
  ffn_kernel<<<MAXTILES, 512, SMEM_BYTES, stream>>>(hidden, gate_p, up_p, down_p,
                                                    rtok, rwgt, texp, out);

  // 6. aux load-balancing loss
  aux_kernel<<<1, 32, 0, stream>>>(cnt, psum, out + (size_t)BT * D);
}